// GNNLayer_71219147702349
// MI455X (gfx1250) — compile-verified
//
#include <hip/hip_runtime.h>
#include <math.h>

// Problem constants
#define NN 50000
#define KK 12
#define DD 64
#define EE 64
#define CC 192   // 2*DD + EE
#define HH 4
#define AA 16
#define EPSV 1e-5f

typedef float v2f __attribute__((ext_vector_type(2)));
typedef float v8f __attribute__((ext_vector_type(8)));

// ---------------- workspace layout (float offsets) ----------------
static const size_t OFF_MEFF   = 0;                                   // NN*KK
static const size_t OFF_PBUF   = OFF_MEFF + (size_t)NN * KK;          // NN*KK*64 (e_lin -> p)
static const size_t OFF_POOLED = OFF_PBUF + (size_t)NN * KK * 64;     // NN*64
static const size_t OFF_PARTA  = OFF_POOLED + (size_t)NN * 64;        // 12500*128
static const size_t OFF_PARTC  = OFF_PARTA + (size_t)12500 * 128;     // 12500*128
static const size_t OFF_PARTD  = OFF_PARTC + (size_t)12500 * 128;     // 12500*128
static const size_t OFF_CNTP   = OFF_PARTD + (size_t)12500 * 128;     // 256
static const size_t OFF_STATS  = OFF_CNTP + 256;

__device__ __forceinline__ float softplus_f(float x) {
    return fmaxf(x, 0.f) + log1pf(__expf(-fabsf(x)));
}

// ---- 16-k-step GEMM segment, 3 M-tiles share each B fragment ----
// B from LDS in k-pair-interleaved layout: sW[(c>>1)*128 + col*2 + (c&1)]
__device__ __forceinline__ void gemm48_lds(const float* __restrict__ A0,
                                           const float* __restrict__ A1,
                                           const float* __restrict__ A2,
                                           const float* sW,
                                           int nl, int kO, int cbase, v8f acc[3][4]) {
    int half = kO >> 1;
#pragma unroll
    for (int t = 0; t < 16; ++t) {
        int col = t * 4 + kO;
        v2f a0 = *(const v2f*)(A0 + col);
        v2f a1 = *(const v2f*)(A1 + col);
        v2f a2 = *(const v2f*)(A2 + col);
        int cpair = ((cbase + t * 4) >> 1) + half;     // (c>>1), c = cbase + t*4 + kO
        const float* wrow = sW + cpair * 128 + nl * 2;
#pragma unroll
        for (int j = 0; j < 4; ++j) {
            v2f b = *(const v2f*)(wrow + j * 32);      // ds_load_b64
            acc[0][j] = __builtin_amdgcn_wmma_f32_16x16x4_f32(
                false, a0, false, b, (short)0, acc[0][j], false, false);
            acc[1][j] = __builtin_amdgcn_wmma_f32_16x16x4_f32(
                false, a1, false, b, (short)0, acc[1][j], false, false);
            acc[2][j] = __builtin_amdgcn_wmma_f32_16x16x4_f32(
                false, a2, false, b, (short)0, acc[2][j], false, false);
        }
    }
}

// B from global, layout W[j*3072 + c*16 + nl]  (W_att1 / W_val: (H,C,A))
__device__ __forceinline__ void gemm48_glb(const float* __restrict__ A0,
                                           const float* __restrict__ A1,
                                           const float* __restrict__ A2,
                                           const float* __restrict__ W,
                                           int nl, int kO, int cbase, v8f acc[3][4]) {
#pragma unroll
    for (int t = 0; t < 16; ++t) {
        int col = t * 4 + kO;
        v2f a0 = *(const v2f*)(A0 + col);
        v2f a1 = *(const v2f*)(A1 + col);
        v2f a2 = *(const v2f*)(A2 + col);
        int c = cbase + col;
#pragma unroll
        for (int j = 0; j < 4; ++j) {
            const float* bp = W + (size_t)j * 3072 + nl;
            v2f b;
            b.x = bp[(size_t)c * 16];
            b.y = bp[(size_t)(c + 1) * 16];
            acc[0][j] = __builtin_amdgcn_wmma_f32_16x16x4_f32(
                false, a0, false, b, (short)0, acc[0][j], false, false);
            acc[1][j] = __builtin_amdgcn_wmma_f32_16x16x4_f32(
                false, a1, false, b, (short)0, acc[1][j], false, false);
            acc[2][j] = __builtin_amdgcn_wmma_f32_16x16x4_f32(
                false, a2, false, b, (short)0, acc[2][j], false, false);
        }
    }
}

// ---------------- P0: effective masks + count partials ----------------
__global__ __launch_bounds__(256) void kprep(const float* __restrict__ masks,
                                             float* __restrict__ meff,
                                             float* __restrict__ cntPart) {
    __shared__ float sred[256];
    int n = blockIdx.x * 256 + threadIdx.x;
    float cn = 0.f;
    if (n < NN) {
        float mv[KK];
        float lens = 0.f;
#pragma unroll
        for (int k = 0; k < KK; ++k) { mv[k] = masks[n * KK + k]; lens += mv[k]; }
        if (lens == 0.f) mv[0] = 1.f;            // fix_row
#pragma unroll
        for (int k = 0; k < KK; ++k) meff[n * KK + k] = mv[k];
        cn = (lens == 0.f) ? 1.f : lens;
    }
    sred[threadIdx.x] = cn;
    __syncthreads();
    for (int s = 128; s > 0; s >>= 1) {
        if (threadIdx.x < s) sred[threadIdx.x] += sred[threadIdx.x + s];
        __syncthreads();
    }
    if (threadIdx.x == 0) cntPart[blockIdx.x] = sred[0];
}

__global__ void kcnt(const float* __restrict__ cntPart, int n, float* __restrict__ statCnt) {
    if (threadIdx.x == 0) {
        float s = 0.f;
        for (int i = 0; i < n; ++i) s += cntPart[i];
        statCnt[0] = s;
    }
}

// ---------------- A: e_lin = c1 @ W_edge + b_edge, masked stats ----------------
// 48 rows per wave (3 M-tiles), W_edge staged in LDS (paired layout, ds_load_b64 B frags)
__global__ __launch_bounds__(128) void kedge_gemm(const float* __restrict__ nodeF,
                                                  const float* __restrict__ edgeF,
                                                  const int* __restrict__ nbrIdx,
                                                  const float* __restrict__ Wedge,
                                                  const float* __restrict__ bedge,
                                                  const float* __restrict__ meff,
                                                  float* __restrict__ eLin,
                                                  float* __restrict__ partA) {
    __shared__ float sW[96 * 128];               // 12288 floats = 48 KB
    for (int idx = threadIdx.x; idx < CC * 64; idx += 128) {
        int c = idx >> 6, col = idx & 63;
        sW[(c >> 1) * 128 + col * 2 + (c & 1)] = Wedge[idx];
    }
    __syncthreads();

    int lane = threadIdx.x & 31, wv = threadIdx.x >> 5;
    int grp = blockIdx.x * 4 + wv;               // 0..12499, 48 rows each
    int G0 = grp * 48;
    int nl = lane & 15, half = lane >> 4, kO = half * 2;

    const float* pS[3];
    const float* pN[3];
    const float* pE[3];
#pragma unroll
    for (int mt = 0; mt < 3; ++mt) {
        int row = G0 + mt * 16 + nl;
        int n = row / KK, k = row % KK;
        pS[mt] = nodeF + (size_t)n * 64;
        pN[mt] = nodeF + (size_t)nbrIdx[n * KK + k] * 64;
        pE[mt] = edgeF + (size_t)row * 64;
    }

    v8f acc[3][4];
#pragma unroll
    for (int j = 0; j < 4; ++j) {
        float bb = bedge[j * 16 + nl];
#pragma unroll
        for (int mt = 0; mt < 3; ++mt)
#pragma unroll
            for (int r = 0; r < 8; ++r) acc[mt][j][r] = bb;
    }

    gemm48_lds(pS[0], pS[1], pS[2], sW, nl, kO, 0,   acc);
    gemm48_lds(pN[0], pN[1], pN[2], sW, nl, kO, 64,  acc);
    gemm48_lds(pE[0], pE[1], pE[2], sW, nl, kO, 128, acc);

    // store e_lin
#pragma unroll
    for (int mt = 0; mt < 3; ++mt)
#pragma unroll
        for (int j = 0; j < 4; ++j)
#pragma unroll
            for (int r = 0; r < 8; ++r)
                eLin[(size_t)(G0 + mt * 16 + half * 8 + r) * 64 + j * 16 + nl] = acc[mt][j][r];

    // masked per-channel sum / sumsq partials
    float mrow[3][8];
#pragma unroll
    for (int mt = 0; mt < 3; ++mt)
#pragma unroll
        for (int r = 0; r < 8; ++r) mrow[mt][r] = meff[G0 + mt * 16 + half * 8 + r];
#pragma unroll
    for (int j = 0; j < 4; ++j) {
        float s0 = 0.f, q0 = 0.f;
#pragma unroll
        for (int mt = 0; mt < 3; ++mt)
#pragma unroll
            for (int r = 0; r < 8; ++r) {
                float x = acc[mt][j][r];
                s0 += x * mrow[mt][r];
                q0 += x * x * mrow[mt][r];
            }
        s0 += __shfl_xor(s0, 16, 32);
        q0 += __shfl_xor(q0, 16, 32);
        if (half == 0) {
            partA[(size_t)grp * 128 + j * 16 + nl] = s0;
            partA[(size_t)grp * 128 + 64 + j * 16 + nl] = q0;
        }
    }
}

// ---------------- finalize a masked BN: partials -> scale/shift ----------------
__global__ void kfinalize(const float* __restrict__ part, int nParts,
                          const float* __restrict__ g, const float* __restrict__ b,
                          const float* __restrict__ cntPtr, float cntConst,
                          float* __restrict__ scale, float* __restrict__ shift) {
    int c = threadIdx.x;
    if (c >= 64) return;
    float cnt = cntPtr ? cntPtr[0] : cntConst;
    float s = 0.f, q = 0.f;
    for (int i = 0; i < nParts; ++i) {
        s += part[(size_t)i * 128 + c];
        q += part[(size_t)i * 128 + 64 + c];
    }
    float mean = s / cnt;
    float var = fmaxf(q / cnt - mean * mean, 0.f);
    float sc = g[c] * rsqrtf(var + EPSV);
    scale[c] = sc;
    shift[c] = b[c] - mean * sc;
}

// ---------------- C: fused edge-update + attention ----------------
__global__ __launch_bounds__(128) void kattn(const float* __restrict__ nodeF,
                                             const float* __restrict__ edgeF,
                                             const int* __restrict__ nbrIdx,
                                             const float* __restrict__ W1,
                                             const float* __restrict__ b1,
                                             const float* __restrict__ W2,
                                             const float* __restrict__ b2v,
                                             const float* __restrict__ Wv,
                                             const float* __restrict__ bv,
                                             const float* __restrict__ meff,
                                             const float* __restrict__ stats,
                                             float* __restrict__ pbuf,      // in: e_lin, out: p
                                             float* __restrict__ outEdge,
                                             float* __restrict__ partC) {
    __shared__ float sEdge[4][48 * 64];   // per-wave edge_updated staging (48 KB)
    __shared__ float sLog[4][48 * 4];     // logits -> att
    __shared__ float sMask[4][48];

    int lane = threadIdx.x & 31, w = threadIdx.x >> 5;
    int group = blockIdx.x * 4 + w;       // 4 nodes per wave
    int R0 = group * 48;
    const float* scaleE = stats + 1;
    const float* shiftE = stats + 65;

    // ---- phase 0: edge BN + softplus, write output + stage in LDS ----
    for (int r = lane; r < 48; r += 32) {
        int row = R0 + r;
        float m = meff[row];
        sMask[w][r] = m;
        const float* el = pbuf + (size_t)row * 64;
        const float* ef = edgeF + (size_t)row * 64;
        float* eo = outEdge + (size_t)row * 64;
        for (int c = 0; c < 64; c += 4) {
            float4 e4 = *(const float4*)(ef + c);
            float4 l4 = *(const float4*)(el + c);
            float4 u;
            u.x = softplus_f(e4.x + (scaleE[c + 0] * l4.x + shiftE[c + 0]) * m);
            u.y = softplus_f(e4.y + (scaleE[c + 1] * l4.y + shiftE[c + 1]) * m);
            u.z = softplus_f(e4.z + (scaleE[c + 2] * l4.z + shiftE[c + 2]) * m);
            u.w = softplus_f(e4.w + (scaleE[c + 3] * l4.w + shiftE[c + 3]) * m);
            *(float4*)(eo + c) = u;
            *(float4*)(&sEdge[w][r * 64 + c]) = u;
        }
    }
    __syncthreads();

    int nl = lane & 15, half = lane >> 4, kO = half * 2;

    const float* pS[3];
    const float* pN[3];
    const float* pE[3];
#pragma unroll
    for (int mt = 0; mt < 3; ++mt) {
        int lrowA = mt * 16 + nl;
        int rowA = R0 + lrowA;
        int n = rowA / KK, kq = rowA % KK;
        pS[mt] = nodeF + (size_t)n * 64;
        pN[mt] = nodeF + (size_t)nbrIdx[n * KK + kq] * 64;
        pE[mt] = &sEdge[w][lrowA * 64];
    }

    // ---- phase 1: h1 = sp(c2 @ W_att1 + b1); logits via cross-lane dot ----
    {
        v8f h[3][4];
#pragma unroll
        for (int j = 0; j < 4; ++j) {
            float bb = b1[j * 16 + nl];
#pragma unroll
            for (int mt = 0; mt < 3; ++mt)
#pragma unroll
                for (int r = 0; r < 8; ++r) h[mt][j][r] = bb;
        }
        gemm48_glb(pS[0], pS[1], pS[2], W1, nl, kO, 0,   h);
        gemm48_glb(pN[0], pN[1], pN[2], W1, nl, kO, 64,  h);
        gemm48_glb(pE[0], pE[1], pE[2], W1, nl, kO, 128, h);

#pragma unroll
        for (int mt = 0; mt < 3; ++mt)
#pragma unroll
            for (int j = 0; j < 4; ++j)
#pragma unroll
                for (int r = 0; r < 8; ++r) h[mt][j][r] = softplus_f(h[mt][j][r]);

        // logits: head j occupies cols j*16..j*16+15 (A == 16)
#pragma unroll
        for (int mt = 0; mt < 3; ++mt)
#pragma unroll
            for (int j = 0; j < 4; ++j) {
                float w2 = W2[j * 16 + nl];
#pragma unroll
                for (int r = 0; r < 8; ++r) {
                    float t = h[mt][j][r] * w2;
                    t += __shfl_xor(t, 1, 32);
                    t += __shfl_xor(t, 2, 32);
                    t += __shfl_xor(t, 4, 32);
                    t += __shfl_xor(t, 8, 32);
                    if (nl == 0)
                        sLog[w][(mt * 16 + half * 8 + r) * 4 + j] = t + b2v[j];
                }
            }
    }
    __syncthreads();

    // ---- phase 2: masked softmax over K per (node, head) ----
    if (lane < 16) {
        int nd = lane >> 2, hh = lane & 3;
        int base = nd * 12;
        float lv[KK];
        float mx = -3.0e38f;
#pragma unroll
        for (int q = 0; q < KK; ++q) {
            float m = sMask[w][base + q];
            float l = (m > 0.f) ? sLog[w][(base + q) * 4 + hh] : -3.0e38f;
            lv[q] = l;
            mx = fmaxf(mx, l);
        }
        float se = 0.f;
#pragma unroll
        for (int q = 0; q < KK; ++q) {
            float e = (lv[q] > -1.0e38f) ? __expf(lv[q] - mx) : 0.f;
            lv[q] = e;
            se += e;
        }
        float inv = 1.f / se;
#pragma unroll
        for (int q = 0; q < KK; ++q) sLog[w][(base + q) * 4 + hh] = lv[q] * inv;
    }
    __syncthreads();

    // ---- phase 3: v = c2 @ W_val + bv ; p = att*v ; masked stats ----
    float ps[4] = {0, 0, 0, 0}, pq[4] = {0, 0, 0, 0};
    {
        v8f v[3][4];
#pragma unroll
        for (int j = 0; j < 4; ++j) {
            float bb = bv[j * 16 + nl];
#pragma unroll
            for (int mt = 0; mt < 3; ++mt)
#pragma unroll
                for (int r = 0; r < 8; ++r) v[mt][j][r] = bb;
        }
        gemm48_glb(pS[0], pS[1], pS[2], Wv, nl, kO, 0,   v);
        gemm48_glb(pN[0], pN[1], pN[2], Wv, nl, kO, 64,  v);
        gemm48_glb(pE[0], pE[1], pE[2], Wv, nl, kO, 128, v);

#pragma unroll
        for (int mt = 0; mt < 3; ++mt)
#pragma unroll
            for (int j = 0; j < 4; ++j)
#pragma unroll
                for (int r = 0; r < 8; ++r) {
                    int lrow = mt * 16 + half * 8 + r;
                    float att = sLog[w][lrow * 4 + j];
                    float m = sMask[w][lrow];
                    float p = att * v[mt][j][r];
                    pbuf[(size_t)(R0 + lrow) * 64 + j * 16 + nl] = p;
                    ps[j] += p * m;
                    pq[j] += p * p * m;
                }
    }
#pragma unroll
    for (int j = 0; j < 4; ++j) {
        ps[j] += __shfl_xor(ps[j], 16, 32);
        pq[j] += __shfl_xor(pq[j], 16, 32);
        if (half == 0) {
            partC[(size_t)group * 128 + j * 16 + nl] = ps[j];
            partC[(size_t)group * 128 + 64 + j * 16 + nl] = pq[j];
        }
    }
}

// ---------------- D: p-BN + softplus + pool over K, out-BN partials ----------------
__global__ __launch_bounds__(256) void kpool(const float* __restrict__ pbuf,
                                             const float* __restrict__ meff,
                                             const float* __restrict__ stats,
                                             float* __restrict__ pooled,
                                             float* __restrict__ partD) {
    __shared__ float sD[256], sQ[256];
    int gid = blockIdx.x * 256 + threadIdx.x;   // exactly NN*64 threads
    int n = gid >> 6, c = gid & 63;
    float sc = stats[129 + c], sh = stats[193 + c];
    float acc = 0.f;
#pragma unroll
    for (int k = 0; k < KK; ++k) {
        float p = pbuf[(size_t)(n * KK + k) * 64 + c];
        float m = meff[n * KK + k];
        acc += softplus_f(sc * p + sh) * m;
    }
    pooled[gid] = acc;
    sD[threadIdx.x] = acc;
    sQ[threadIdx.x] = acc * acc;
    __syncthreads();
    if (threadIdx.x < 64) {
        float s = 0.f, q = 0.f;
#pragma unroll
        for (int i = 0; i < 4; ++i) {
            s += sD[threadIdx.x + 64 * i];
            q += sQ[threadIdx.x + 64 * i];
        }
        partD[(size_t)blockIdx.x * 128 + threadIdx.x] = s;
        partD[(size_t)blockIdx.x * 128 + 64 + threadIdx.x] = q;
    }
}

// ---------------- F: residual + out BN ----------------
__global__ __launch_bounds__(256) void kresid(const float* __restrict__ nodeF,
                                              const float* __restrict__ pooled,
                                              const float* __restrict__ stats,
                                              float* __restrict__ outNode) {
    int gid = blockIdx.x * 256 + threadIdx.x;   // exactly NN*64 threads
    int c = gid & 63;
    outNode[gid] = nodeF[gid] + stats[257 + c] * pooled[gid] + stats[321 + c];
}

extern "C" void kernel_launch(void* const* d_in, const int* in_sizes, int n_in,
                              void* d_out, int out_size, void* d_ws, size_t ws_size,
                              hipStream_t stream) {
    (void)in_sizes; (void)n_in; (void)out_size; (void)ws_size;
    const float* nodeF = (const float*)d_in[0];
    const float* edgeF = (const float*)d_in[1];
    const int*   nbrIdx = (const int*)d_in[2];
    const float* masks = (const float*)d_in[3];
    const float* Wedge = (const float*)d_in[4];
    const float* bedge = (const float*)d_in[5];
    const float* gE = (const float*)d_in[6];
    const float* bE = (const float*)d_in[7];
    const float* W1 = (const float*)d_in[8];
    const float* b1 = (const float*)d_in[9];
    const float* W2 = (const float*)d_in[10];
    const float* b2 = (const float*)d_in[11];
    const float* Wv = (const float*)d_in[12];
    const float* bv = (const float*)d_in[13];
    const float* gA = (const float*)d_in[14];
    const float* bA = (const float*)d_in[15];
    const float* gO = (const float*)d_in[16];
    const float* bO = (const float*)d_in[17];

    float* out = (float*)d_out;
    float* outNode = out;
    float* outEdge = out + (size_t)NN * DD;

    float* ws = (float*)d_ws;
    float* meff    = ws + OFF_MEFF;
    float* pbuf    = ws + OFF_PBUF;
    float* pooled  = ws + OFF_POOLED;
    float* partA   = ws + OFF_PARTA;
    float* partC   = ws + OFF_PARTC;
    float* partD   = ws + OFF_PARTD;
    float* cntPart = ws + OFF_CNTP;
    float* stats   = ws + OFF_STATS;

    const int p0Blocks = (NN + 255) / 256;      // 196

    kprep<<<p0Blocks, 256, 0, stream>>>(masks, meff, cntPart);
    kcnt<<<1, 32, 0, stream>>>(cntPart, p0Blocks, stats);

    kedge_gemm<<<3125, 128, 0, stream>>>(nodeF, edgeF, nbrIdx, Wedge, bedge,
                                         meff, pbuf, partA);
    kfinalize<<<1, 64, 0, stream>>>(partA, 12500, gE, bE, stats, 0.f,
                                    stats + 1, stats + 65);

    kattn<<<3125, 128, 0, stream>>>(nodeF, edgeF, nbrIdx, W1, b1, W2, b2,
                                    Wv, bv, meff, stats, pbuf, outEdge, partC);
    kfinalize<<<1, 64, 0, stream>>>(partC, 12500, gA, bA, stats, 0.f,
                                    stats + 129, stats + 193);

    kpool<<<12500, 256, 0, stream>>>(pbuf, meff, stats, pooled, partD);
    kfinalize<<<1, 64, 0, stream>>>(partD, 12500, gO, bO, nullptr, (float)NN,
                                    stats + 257, stats + 321);

    kresid<<<12500, 256, 0, stream>>>(nodeF, pooled, stats, outNode);
}